// GraphFeatureGenerator_44659069943979
// MI455X (gfx1250) — compile-verified
//
#include <hip/hip_runtime.h>
#include <hip/hip_bf16.h>
#include <stdint.h>

// GraphFeatureGenerator for MI455X (gfx1250).
// Memory/atomic-bound scatter-mean. CDNA5 paths used:
//  - GLOBAL_LOAD_ASYNC_TO_LDS_B128 (ASYNCcnt DMA) to stage edge-index tiles
//  - S_WAIT_ASYNCCNT split counter
//  - global_atomic_add_f32 hardware float atomics (L2-resident working set)
//  - wave32-aware 16-lane edge groups (2 groups per wave), b128 row gathers

#define DIM     64
#define TILE    256   // edges staged per block iteration
#define GLANES  16    // lanes per edge group (each lane owns 4 dims)
#define BLOCK   256   // 8 waves, 16 groups

// ---- CDNA5 async-DMA helpers -------------------------------------------

__device__ __forceinline__ void cdna5_wait_async0() {
#if __has_builtin(__builtin_amdgcn_s_wait_asynccnt)
    __builtin_amdgcn_s_wait_asynccnt(0);
#else
    asm volatile("s_wait_asynccnt 0" ::: "memory");
#endif
}

// GLOBAL_LOAD_ASYNC_TO_LDS_B128: VDST = LDS byte address (VGPR),
// VADDR = 64-bit global address, saddr = off. Tracked by ASYNCcnt.
__device__ __forceinline__ void cdna5_async_b128_to_lds(const void* gsrc,
                                                        void* lds_dst) {
    uint32_t lds_off = (uint32_t)(uintptr_t)lds_dst;   // low 32 bits = LDS offset
    uint64_t gaddr   = (uint64_t)(uintptr_t)gsrc;
    asm volatile("global_load_async_to_lds_b128 %0, %1, off"
                 :: "v"(lds_off), "v"(gaddr)
                 : "memory");
}

// ---- hardware f32 atomic add (global_atomic_add_f32, no CAS loop) ------
__device__ __forceinline__ void hw_atomic_add(float* p, float v) {
#if defined(__HIP_PLATFORM_AMD__) || defined(__AMDGCN__)
    unsafeAtomicAdd(p, v);
#else
    atomicAdd(p, v);
#endif
}

// Stage one TILE of edge indices into LDS (async DMA fast path).
__device__ __forceinline__ void stage_tile(const int* __restrict__ eu,
                                           const int* __restrict__ ei,
                                           int base, int num_edges, int tid,
                                           int* s_u, int* s_i) {
    if (base + TILE <= num_edges) {
        // 128 threads each DMA 16 B: 64 cover s_u, 64 cover s_i.
        if (tid < 64) {
            cdna5_async_b128_to_lds(eu + base + tid * 4, &s_u[tid * 4]);
        } else if (tid < 128) {
            const int t = tid - 64;
            cdna5_async_b128_to_lds(ei + base + t * 4, &s_i[t * 4]);
        }
    } else {
        const int e = base + tid;
        if (e < num_edges) {
            s_u[tid] = eu[e];
            s_i[tid] = ei[e];
        }
    }
    cdna5_wait_async0();
    __syncthreads();
}

// ---- Pass 1: per edge e=(u,i):
//   out_nb[u,:]   += item_emb[i,:]
//   item_sum[i,:] += user_emb[u,:]
//   ucnt[u] += 1 ; icnt[i] += 1
__global__ __launch_bounds__(BLOCK)
void pass1_kernel(const float* __restrict__ uemb, const float* __restrict__ iemb,
                  const int* __restrict__ eu, const int* __restrict__ ei,
                  float* __restrict__ out_nb, float* __restrict__ item_sum,
                  float* __restrict__ ucnt, float* __restrict__ icnt,
                  int num_edges) {
    __shared__ int s_u[TILE];
    __shared__ int s_i[TILE];
    const int tid  = threadIdx.x;
    const int lane = tid & (GLANES - 1);
    const int grp  = tid >> 4;          // 0..15

    for (int base = blockIdx.x * TILE; base < num_edges; base += gridDim.x * TILE) {
        stage_tile(eu, ei, base, num_edges, tid, s_u, s_i);

        #pragma unroll 1
        for (int k = 0; k < GLANES; ++k) {
            const int eg = base + grp * GLANES + k;
            if (eg >= num_edges) break;
            const int u = s_u[grp * GLANES + k];   // LDS broadcast
            const int i = s_i[grp * GLANES + k];

            const float4 iv = *(const float4*)(iemb + (size_t)i * DIM + lane * 4);
            const float4 uv = *(const float4*)(uemb + (size_t)u * DIM + lane * 4);

            float* on = out_nb + (size_t)u * DIM + lane * 4;
            hw_atomic_add(on + 0, iv.x); hw_atomic_add(on + 1, iv.y);
            hw_atomic_add(on + 2, iv.z); hw_atomic_add(on + 3, iv.w);

            float* is = item_sum + (size_t)i * DIM + lane * 4;
            hw_atomic_add(is + 0, uv.x); hw_atomic_add(is + 1, uv.y);
            hw_atomic_add(is + 2, uv.z); hw_atomic_add(is + 3, uv.w);

            if (lane == 0) {
                hw_atomic_add(ucnt + u, 1.0f);
                hw_atomic_add(icnt + i, 1.0f);
            }
        }
        __syncthreads();
    }
}

// ---- item_sum -> item_mean in place: /= max(icnt, 1)
__global__ __launch_bounds__(BLOCK)
void item_mean_kernel(float* __restrict__ item_sum, const float* __restrict__ icnt,
                      int n) {
    for (int idx = blockIdx.x * blockDim.x + threadIdx.x; idx < n;
         idx += gridDim.x * blockDim.x) {
        const float c = fmaxf(icnt[idx >> 6], 1.0f);
        item_sum[idx] = item_sum[idx] / c;
    }
}

// ---- Pass 2: per edge e=(u,i): out_co[u,:] += item_mean[i,:]
__global__ __launch_bounds__(BLOCK)
void pass2_kernel(const float* __restrict__ item_mean,
                  const int* __restrict__ eu, const int* __restrict__ ei,
                  float* __restrict__ out_co, int num_edges) {
    __shared__ int s_u[TILE];
    __shared__ int s_i[TILE];
    const int tid  = threadIdx.x;
    const int lane = tid & (GLANES - 1);
    const int grp  = tid >> 4;

    for (int base = blockIdx.x * TILE; base < num_edges; base += gridDim.x * TILE) {
        stage_tile(eu, ei, base, num_edges, tid, s_u, s_i);

        #pragma unroll 1
        for (int k = 0; k < GLANES; ++k) {
            const int eg = base + grp * GLANES + k;
            if (eg >= num_edges) break;
            const int u = s_u[grp * GLANES + k];
            const int i = s_i[grp * GLANES + k];

            const float4 mv = *(const float4*)(item_mean + (size_t)i * DIM + lane * 4);
            float* oc = out_co + (size_t)u * DIM + lane * 4;
            hw_atomic_add(oc + 0, mv.x); hw_atomic_add(oc + 1, mv.y);
            hw_atomic_add(oc + 2, mv.z); hw_atomic_add(oc + 3, mv.w);
        }
        __syncthreads();
    }
}

// ---- Finalize: both user-side accumulators /= max(ucnt, 1)
__global__ __launch_bounds__(BLOCK)
void finalize_kernel(float* __restrict__ out_nb, float* __restrict__ out_co,
                     const float* __restrict__ ucnt, int n) {
    for (int idx = blockIdx.x * blockDim.x + threadIdx.x; idx < n;
         idx += gridDim.x * blockDim.x) {
        const float c = fmaxf(ucnt[idx >> 6], 1.0f);
        out_nb[idx] = out_nb[idx] / c;
        out_co[idx] = out_co[idx] / c;
    }
}

extern "C" void kernel_launch(void* const* d_in, const int* in_sizes, int n_in,
                              void* d_out, int out_size, void* d_ws, size_t ws_size,
                              hipStream_t stream) {
    const float* uemb = (const float*)d_in[0];
    const float* iemb = (const float*)d_in[1];
    const int*   eu   = (const int*)d_in[2];
    const int*   ei   = (const int*)d_in[3];

    const int num_user  = in_sizes[0] / DIM;   // 100000
    const int num_item  = in_sizes[1] / DIM;   // 50000
    const int num_edges = in_sizes[2];         // 3200000

    float* out_nb = (float*)d_out;                       // [num_user, 64]
    float* out_co = out_nb + (size_t)num_user * DIM;     // [num_user, 64]

    // workspace layout: item_sum | ucnt | icnt
    float* item_sum = (float*)d_ws;
    float* ucnt     = item_sum + (size_t)num_item * DIM;
    float* icnt     = ucnt + num_user;
    const size_t ws_floats = (size_t)num_item * DIM + num_user + num_item;

    // deterministic zero-init every call (graph-capturable memset nodes)
    hipMemsetAsync(d_out, 0, (size_t)out_size * sizeof(float), stream);
    hipMemsetAsync(d_ws, 0, ws_floats * sizeof(float), stream);

    const int tiles = (num_edges + TILE - 1) / TILE;     // 12500 blocks

    pass1_kernel<<<tiles, BLOCK, 0, stream>>>(uemb, iemb, eu, ei,
                                              out_nb, item_sum, ucnt, icnt,
                                              num_edges);

    const int n_item_elems = num_item * DIM;
    item_mean_kernel<<<(n_item_elems + BLOCK - 1) / BLOCK, BLOCK, 0, stream>>>(
        item_sum, icnt, n_item_elems);

    pass2_kernel<<<tiles, BLOCK, 0, stream>>>(item_sum, eu, ei, out_co, num_edges);

    const int n_user_elems = num_user * DIM;
    finalize_kernel<<<(n_user_elems + BLOCK - 1) / BLOCK, BLOCK, 0, stream>>>(
        out_nb, out_co, ucnt, n_user_elems);
}